// HingeRankLoss_39041252721038
// MI455X (gfx1250) — compile-verified
//
#include <hip/hip_runtime.h>
#include <hip/hip_bf16.h>

#define MARGIN 0.1f

typedef __attribute__((ext_vector_type(2))) float v2f;
typedef __attribute__((ext_vector_type(8))) float v8f;

// Exact fp32 sum of all 32 lanes of a wave, using one V_WMMA_F32_16X16X4_F32.
// A = {v,0} per lane  -> A[m][k] holds v_m (k=0, lanes 0-15) and v_{m+16}
// (k=2, lanes 16-31), zeros elsewhere.  B = all ones.
// D[m][n] = v_m + v_{m+16}.  Lane l<16 holds D[0..7][l] in its 8 acc VGPRs,
// lane l>=16 holds D[8..15][l-16]; summing the 8 components and xor-16
// exchanging yields the full 32-lane sum in every lane (pure fp32 RNE tree).
__device__ __forceinline__ float wave_sum32(float v) {
#if __has_builtin(__builtin_amdgcn_wmma_f32_16x16x4_f32)
    v2f a = {v, 0.0f};
    v2f b = {1.0f, 1.0f};
    v8f c = {};
    v8f d = __builtin_amdgcn_wmma_f32_16x16x4_f32(
        /*neg_a=*/false, a, /*neg_b=*/false, b,
        /*c_mod=*/(short)0, c, /*reuse_a=*/false, /*reuse_b=*/false);
    float s = ((d[0] + d[1]) + (d[2] + d[3])) + ((d[4] + d[5]) + (d[6] + d[7]));
    s += __shfl_xor(s, 16);
    return s;
#else
    #pragma unroll
    for (int off = 16; off > 0; off >>= 1) v += __shfl_xor(v, off);
    return v;
#endif
}

// One 256-thread block (8 wave32) per row; L assumed 2048 (8 elems/thread).
__global__ __launch_bounds__(256) void hinge_row_kernel(
    const float* __restrict__ scores,
    const long long* __restrict__ lens,
    const long long* __restrict__ labels,
    float2* __restrict__ rowout,
    int L)
{
    const int row = blockIdx.x;
    const int tid = threadIdx.x;
    const int len = (int)lens[row];

    const float4*    s4 = (const float4*)   (scores + (size_t)row * (size_t)L);
    const longlong2* l2 = (const longlong2*)(labels + (size_t)row * (size_t)L);

    float    sv[8];        // row chunk stays register-resident: single HBM pass
    unsigned negbits = 0;  // valid-negative mask for my 8 elements
    float pos_p = 0.0f, pos_c = 0.0f;

    #pragma unroll
    for (int t = 0; t < 2; ++t) {              // 2 * 256 float4 = 2048 elems
        const int u  = tid + (t << 8);         // coalesced float4 index
        const int j0 = u << 2;
        float4    s  = s4[u];
        longlong2 la = l2[2 * u];
        longlong2 lb = l2[2 * u + 1];
        float     ss[4] = { s.x, s.y, s.z, s.w };
        long long lv[4] = { la.x, la.y, lb.x, lb.y };
        #pragma unroll
        for (int k = 0; k < 4; ++k) {
            const int  j     = j0 + k;
            const bool valid = (j < len);
            const bool isp   = (lv[k] != 0);
            sv[(t << 2) + k] = ss[k];
            if (valid && isp)  { pos_p += ss[k]; pos_c += 1.0f; }
            if (valid && !isp) { negbits |= (1u << ((t << 2) + k)); }
        }
    }

    // ---- phase 1: positive score / has_pos (WMMA wave reduce + LDS) ----
    pos_p = wave_sum32(pos_p);
    pos_c = wave_sum32(pos_c);

    __shared__ float2 red[8];
    const int w = tid >> 5, l = tid & 31;
    if (l == 0) red[w] = make_float2(pos_p, pos_c);
    __syncthreads();

    float PP = 0.0f, PC = 0.0f;
    #pragma unroll
    for (int i = 0; i < 8; ++i) { PP += red[i].x; PC += red[i].y; }
    const float chosen = (PC > 0.0f) ? PP : -MARGIN;
    __syncthreads();   // red[] is reused below

    // ---- phase 2: hinge over valid negatives (from registers, no reload) ----
    float nsum = 0.0f, ncnt = 0.0f;
    #pragma unroll
    for (int e = 0; e < 8; ++e) {
        if (negbits & (1u << e)) {
            nsum += fmaxf(MARGIN + sv[e] - chosen, 0.0f);
            ncnt += 1.0f;
        }
    }
    nsum = wave_sum32(nsum);
    ncnt = wave_sum32(ncnt);
    if (l == 0) red[w] = make_float2(nsum, ncnt);
    __syncthreads();

    if (tid == 0) {
        float NS = 0.0f, NC = 0.0f;
        #pragma unroll
        for (int i = 0; i < 8; ++i) { NS += red[i].x; NC += red[i].y; }
        const bool  valid_obs  = (len > 0) && (NC > 0.0f);
        const float per_sample = NS / fmaxf(NC, 1.0f);
        rowout[row] = valid_obs ? make_float2(per_sample, 1.0f)
                                : make_float2(0.0f, 0.0f);
    }
}

// Deterministic single-block reduction of 32768 (per_sample, flag) pairs.
__global__ __launch_bounds__(256) void hinge_finalize_kernel(
    const float2* __restrict__ rowout, float* __restrict__ out, int nrows)
{
    const int tid = threadIdx.x;
    float s = 0.0f, c = 0.0f;
    for (int i = tid; i < nrows; i += 256) {
        float2 v = rowout[i];
        s += v.x; c += v.y;
    }
    s = wave_sum32(s);
    c = wave_sum32(c);

    __shared__ float2 red[8];
    const int w = tid >> 5, l = tid & 31;
    if (l == 0) red[w] = make_float2(s, c);
    __syncthreads();

    if (tid == 0) {
        float S = 0.0f, C = 0.0f;
        #pragma unroll
        for (int i = 0; i < 8; ++i) { S += red[i].x; C += red[i].y; }
        out[0] = (C > 0.0f) ? (S / fmaxf(C, 1.0f)) : 0.0f;
    }
}

extern "C" void kernel_launch(void* const* d_in, const int* in_sizes, int n_in,
                              void* d_out, int out_size, void* d_ws, size_t ws_size,
                              hipStream_t stream) {
    const float*     scores = (const float*)d_in[0];
    const long long* lens   = (const long long*)d_in[1];
    const long long* labels = (const long long*)d_in[2];
    const int Bn = in_sizes[1];            // 32768 rows
    const int L  = in_sizes[0] / Bn;       // 2048 cols

    float2* rowout = (float2*)d_ws;        // Bn * 8 bytes of scratch

    hinge_row_kernel<<<Bn, 256, 0, stream>>>(scores, lens, labels, rowout, L);
    hinge_finalize_kernel<<<1, 256, 0, stream>>>(rowout, (float*)d_out, Bn);
}